// MRConv2d_66623532696234
// MI455X (gfx1250) — compile-verified
//
#include <hip/hip_runtime.h>
#include <math.h>

#define B_    8
#define C_    192
#define N_    3136
#define K_    9
#define OC    192
#define CIN   384          // 2*C
#define MTOT  (B_ * N_)    // 25088
#define LDH   384
#define MTILE 128
#define RPB   96           // output rows per block (2 block groups cover 192)
#define RT    6            // 16-row tiles per wave
#define PAD   36           // LDS row pitch (floats): 16B-aligned rows, conflict-free
#define BN_EPS 1e-5f

typedef float v2f __attribute__((ext_vector_type(2)));
typedef float v8f __attribute__((ext_vector_type(8)));

// ---------------------------------------------------------------------------
// 1) Tiled transpose (B,C,N) -> (B,N,C) so gathers read channel-contiguous.
// ---------------------------------------------------------------------------
__global__ __launch_bounds__(256)
void transpose_bcn_to_bnc(const float* __restrict__ src, float* __restrict__ dst,
                          int ldDst) {
  __shared__ float tile[32][33];
  const int b  = blockIdx.z;
  const int n0 = blockIdx.x * 32;
  const int c0 = blockIdx.y * 32;
  const int tx = threadIdx.x, ty = threadIdx.y;   // block (32, 8)
  const float* s = src + (size_t)b * C_ * N_;
#pragma unroll
  for (int j = 0; j < 32; j += 8)
    tile[ty + j][tx] = s[(size_t)(c0 + ty + j) * N_ + (n0 + tx)];
  __syncthreads();
  float* d = dst + (size_t)b * N_ * ldDst;
#pragma unroll
  for (int j = 0; j < 32; j += 8)
    d[(size_t)(n0 + ty + j) * ldDst + (c0 + tx)] = tile[tx][ty + j];
}

// ---------------------------------------------------------------------------
// 2) rel[c] = max_k ( yT[b, src_k, c] - xT[b, dst_k, c] ) -> H[:, 192:384].
// ---------------------------------------------------------------------------
__global__ __launch_bounds__(192)
void gather_maxrel(float* H,                               // [MTOT][384], rw
                   const float* __restrict__ yT,           // [MTOT][192]
                   const long long* __restrict__ edge) {   // (2,B,N,K) int64
  const int m = blockIdx.x;
  const int b = m / N_;
  const int n = m % N_;
  __shared__ int srcI[K_], dstI[K_];
  const int t = threadIdx.x;
  if (t < K_) {
    srcI[t] = (int)edge[(((size_t)0 * B_ + b) * N_ + n) * K_ + t];
    dstI[t] = (int)edge[(((size_t)1 * B_ + b) * N_ + n) * K_ + t];
  }
  __syncthreads();
  const int c = t;
  float r = -INFINITY;
#pragma unroll
  for (int k = 0; k < K_; ++k) {
    float xv = H [((size_t)b * N_ + dstI[k]) * LDH + c];
    float yv = yT[((size_t)b * N_ + srcI[k]) * C_  + c];
    r = fmaxf(r, yv - xv);
  }
  H[(size_t)m * LDH + C_ + c] = r;
}

// ---------------------------------------------------------------------------
// 3) GEMM via V_WMMA_F32_16X16X4_F32.
//    Grid (196, 2): block = 96 D-rows x 128 D-cols, 8 waves, 48 wmma/chunk.
//    Direct b128 global->LDS staging (no reg double-buffer: avoids spills,
//    leaves VGPR headroom so the scheduler pipelines ds_load_2addr_b64).
// ---------------------------------------------------------------------------
__global__ __launch_bounds__(256)
void gemm_wmma(const float* __restrict__ Hmat,   // [MTOT][384]
               const float* __restrict__ W,      // [192][384] row-major
               float* __restrict__ out) {        // (B, OC, N) layout
  __shared__ __align__(16) float Ws[RPB][PAD];     // 96 x 32 chunk of W
  __shared__ __align__(16) float Hs[MTILE][PAD];   // 128 x 32 chunk of H

  const int tid  = threadIdx.x;
  const int wave = tid >> 5;
  const int lane = tid & 31;
  const int half = lane >> 4;     // 0: K={kk,kk+1}, 1: K={kk+2,kk+3}
  const int lrow = lane & 15;     // A: M row; B: N column
  const int mblock  = blockIdx.x * MTILE;
  const int rowBase = blockIdx.y * RPB;

  v8f acc[RT];
#pragma unroll
  for (int r = 0; r < RT; ++r) { v8f z = {}; acc[r] = z; }

  for (int chunk = 0; chunk < CIN / 32; ++chunk) {
    const int c0 = chunk * 32;
    __syncthreads();            // previous compute done reading LDS
#pragma unroll
    for (int j = 0; j < 3; ++j) {         // W chunk: 96 rows x 8 float4
      int i = tid + j * 256, o = i >> 3, q = i & 7;
      float4 v = *(const float4*)&W[(size_t)(rowBase + o) * CIN + c0 + q * 4];
      *(float4*)&Ws[o][q * 4] = v;
    }
#pragma unroll
    for (int j = 0; j < 4; ++j) {         // H chunk: 128 rows x 8 float4
      int i = tid + j * 256, r = i >> 3, q = i & 7;
      float4 v = *(const float4*)&Hmat[(size_t)(mblock + r) * LDH + c0 + q * 4];
      *(float4*)&Hs[r][q * 4] = v;
    }
    __syncthreads();            // staged data visible

#pragma unroll
    for (int kk = 0; kk < 32; kk += 4) {
      v2f bfrag = *(const v2f*)&Hs[wave * 16 + lrow][kk + 2 * half];
#pragma unroll
      for (int r = 0; r < RT; ++r) {
        v2f afrag = *(const v2f*)&Ws[r * 16 + lrow][kk + 2 * half];
        acc[r] = __builtin_amdgcn_wmma_f32_16x16x4_f32(
            false, afrag, false, bfrag, (short)0, acc[r], false, false);
      }
    }
  }

  // D layout: VGPR v -> row M=v (+8 for lane>=16), lanes -> N columns.
  // m-tiles never straddle a batch boundary (N % 16 == 0).
  const int m  = mblock + wave * 16 + lrow;
  const int bb = m / N_;
  const int nn = m % N_;
#pragma unroll
  for (int r = 0; r < RT; ++r) {
#pragma unroll
    for (int v = 0; v < 8; ++v) {
      int o = rowBase + r * 16 + v + 8 * half;
      out[((size_t)bb * OC + o) * N_ + nn] = acc[r][v];
    }
  }
}

// ---------------------------------------------------------------------------
// 4) Per-channel batch statistics over (B, N): mean + rstd
// ---------------------------------------------------------------------------
__global__ __launch_bounds__(256)
void bn_stats(const float* __restrict__ out, float* __restrict__ stats) {
  const int o = blockIdx.x;
  float s = 0.f, sq = 0.f;
  for (int b = 0; b < B_; ++b) {
    const float* p = out + ((size_t)b * OC + o) * N_;
    for (int n = threadIdx.x; n < N_; n += 256) {
      float v = p[n];
      s += v; sq += v * v;
    }
  }
  __shared__ float ssum[256], ssq[256];
  ssum[threadIdx.x] = s; ssq[threadIdx.x] = sq;
  __syncthreads();
  for (int st = 128; st > 0; st >>= 1) {
    if (threadIdx.x < st) {
      ssum[threadIdx.x] += ssum[threadIdx.x + st];
      ssq [threadIdx.x] += ssq [threadIdx.x + st];
    }
    __syncthreads();
  }
  if (threadIdx.x == 0) {
    float mean = ssum[0] / (float)MTOT;
    float var  = ssq[0]  / (float)MTOT - mean * mean;
    stats[o]      = mean;
    stats[OC + o] = rsqrtf(var + BN_EPS);
  }
}

// ---------------------------------------------------------------------------
// 5) In-place BN affine + exact GELU
// ---------------------------------------------------------------------------
__global__ __launch_bounds__(256)
void bn_gelu(float* __restrict__ out, const float* __restrict__ stats,
             const float* __restrict__ gamma, const float* __restrict__ beta) {
  size_t i = (size_t)blockIdx.x * 256 + threadIdx.x;
  const size_t total = (size_t)B_ * OC * N_;
  if (i >= total) return;
  int o = (int)((i / N_) % OC);
  float v = out[i];
  float yn = (v - stats[o]) * stats[OC + o] * gamma[o] + beta[o];
  out[i] = 0.5f * yn * (1.0f + erff(yn * 0.70710678118654752f));
}

// ---------------------------------------------------------------------------
extern "C" void kernel_launch(void* const* d_in, const int* in_sizes, int n_in,
                              void* d_out, int out_size, void* d_ws, size_t ws_size,
                              hipStream_t stream) {
  const float*     x     = (const float*)d_in[0];
  const float*     y     = (const float*)d_in[1];
  const long long* edge  = (const long long*)d_in[2];
  const float*     W     = (const float*)d_in[3];
  const float*     gamma = (const float*)d_in[4];
  const float*     beta  = (const float*)d_in[5];
  float*           out   = (float*)d_out;

  // workspace layout
  float* H     = (float*)d_ws;                  // MTOT * 384 floats
  float* yT    = H  + (size_t)MTOT * LDH;       // MTOT * 192 floats
  float* stats = yT + (size_t)MTOT * C_;        // 2 * OC floats

  dim3 tb(32, 8);
  dim3 tg(N_ / 32, C_ / 32, B_);
  transpose_bcn_to_bnc<<<tg, tb, 0, stream>>>(x, H, LDH);   // x -> H[:, 0:192]
  transpose_bcn_to_bnc<<<tg, tb, 0, stream>>>(y, yT, C_);   // y -> yT

  gather_maxrel<<<MTOT, 192, 0, stream>>>(H, yT, edge);     // H[:, 192:384]

  dim3 gg(MTOT / MTILE, 2);                                 // 392 blocks
  gemm_wmma<<<gg, 256, 0, stream>>>(H, W, out);

  bn_stats<<<OC, 256, 0, stream>>>(out, stats);

  size_t total = (size_t)B_ * OC * N_;
  bn_gelu<<<(unsigned)((total + 255) / 256), 256, 0, stream>>>(out, stats, gamma, beta);
}